// GAT_67577015435453
// MI455X (gfx1250) — compile-verified
//
#include <hip/hip_runtime.h>
#include <hip/hip_bf16.h>

typedef __attribute__((ext_vector_type(2))) float v2f;
typedef __attribute__((ext_vector_type(8))) float v8f;

// Address-space-qualified pointers: force global_load_* / ds_load_* instead of
// flat_* (flat charges both LOADcnt and DScnt on CDNA5).
typedef const __attribute__((address_space(1))) float* gcp;
typedef __attribute__((address_space(1))) float*       gp;
typedef const __attribute__((address_space(1))) int*   gip;
typedef const __attribute__((address_space(3))) float* lcp;
typedef __attribute__((address_space(3))) float*       lp;
typedef const __attribute__((address_space(1))) v2f*   gv2;
typedef const __attribute__((address_space(3))) v2f*   lv2;

constexpr int Bc = 8, Nc = 512, Dc = 128, Hc = 64;
constexpr float kSlope = 0.2f;
constexpr float kNeg = -3.0e38f;

__device__ __forceinline__ float lrelu(float x) { return x > 0.0f ? x : kSlope * x; }

// C(16x64) += A(16xK) * W(Kx64) via V_WMMA_F32_16X16X4_F32.
// A from global memory.
template <int K>
__device__ __forceinline__ void wave_gemm_g(gcp A, int lda, gcp W, v8f acc[4]) {
  const int lane = threadIdx.x & 31;
  const int l15  = lane & 15;
  const int koff = (lane >> 4) * 2;
#pragma unroll 4
  for (int k0 = 0; k0 < K; k0 += 4) {
    const int ka = k0 + koff;
    v2f a = *(gv2)&A[l15 * lda + ka];            // global_load_b64
#pragma unroll
    for (int nt = 0; nt < 4; ++nt) {
      const int n = nt * 16 + l15;
      v2f bf;
      bf.x = W[ka * Hc + n];
      bf.y = W[(ka + 1) * Hc + n];
      acc[nt] = __builtin_amdgcn_wmma_f32_16x16x4_f32(false, a, false, bf,
                                                      (short)0, acc[nt], false, false);
    }
  }
}

// Same, A staged in LDS.
template <int K>
__device__ __forceinline__ void wave_gemm_l(lcp A, int lda, gcp W, v8f acc[4]) {
  const int lane = threadIdx.x & 31;
  const int l15  = lane & 15;
  const int koff = (lane >> 4) * 2;
#pragma unroll 4
  for (int k0 = 0; k0 < K; k0 += 4) {
    const int ka = k0 + koff;
    v2f a = *(lv2)&A[l15 * lda + ka];            // ds_load_b64
#pragma unroll
    for (int nt = 0; nt < 4; ++nt) {
      const int n = nt * 16 + l15;
      v2f bf;
      bf.x = W[ka * Hc + n];
      bf.y = W[(ka + 1) * Hc + n];
      acc[nt] = __builtin_amdgcn_wmma_f32_16x16x4_f32(false, a, false, bf,
                                                      (short)0, acc[nt], false, false);
    }
  }
}

__device__ __forceinline__ void bias_act(v8f acc[4], gcp bias, bool act) {
  const int l15 = threadIdx.x & 15;
#pragma unroll
  for (int nt = 0; nt < 4; ++nt) {
    const float bv = bias[nt * 16 + l15];
#pragma unroll
    for (int r = 0; r < 8; ++r) {
      float t = acc[nt][r] + bv;
      acc[nt][r] = act ? lrelu(t) : t;
    }
  }
}

// Scatter a C/D tile (4 n-tiles of 16x16) to row-major 16x64.
__device__ __forceinline__ void store_tile_l(lp dst, int ldd, const v8f acc[4]) {
  const int lane  = threadIdx.x & 31;
  const int l15   = lane & 15;
  const int mbase = (lane >> 4) * 8;
#pragma unroll
  for (int nt = 0; nt < 4; ++nt)
#pragma unroll
    for (int r = 0; r < 8; ++r)
      dst[(mbase + r) * ldd + nt * 16 + l15] = acc[nt][r];
}

__device__ __forceinline__ void store_tile_g(gp dst, int ldd, const v8f acc[4]) {
  const int lane  = threadIdx.x & 31;
  const int l15   = lane & 15;
  const int mbase = (lane >> 4) * 8;
#pragma unroll
  for (int nt = 0; nt < 4; ++nt)
#pragma unroll
    for (int r = 0; r < 8; ++r)
      dst[(mbase + r) * ldd + nt * 16 + l15] = acc[nt][r];
}

// ---------------- Kernel A: both MLPs + score projections --------------------
__global__ void __launch_bounds__(256)
gat_mlp_kernel(const float* __restrict__ x,
               const float* __restrict__ self_w1, const float* __restrict__ self_b1,
               const float* __restrict__ self_w2, const float* __restrict__ self_b2,
               const float* __restrict__ nb_w1,  const float* __restrict__ nb_b1,
               const float* __restrict__ nb_w2,  const float* __restrict__ nb_b2,
               const float* __restrict__ comb_w1, const float* __restrict__ comb_b1,
               float* __restrict__ self_e, float* __restrict__ nb_e,
               float* __restrict__ u, float* __restrict__ v) {
  constexpr int LDT = 68;                      // pad to dodge 64-bank conflicts
  __shared__ float tiles[8 * 16 * LDT];
  const int wave = threadIdx.x >> 5;
  lp wt = (lp)(tiles + wave * 16 * LDT);
  const int tile = blockIdx.x * 8 + wave;      // 256 tiles of 16 rows
  const int row0 = tile * 16;

  const float* W1[2] = {self_w1, nb_w1};
  const float* B1[2] = {self_b1, nb_b1};
  const float* W2[2] = {self_w2, nb_w2};
  const float* B2[2] = {self_b2, nb_b2};
  float*       EO[2] = {self_e, nb_e};
  const float* CW[2] = {comb_w1, comb_w1 + Hc * Hc};  // w1_s / w1_n
  float*       UV[2] = {u, v};

  for (int br = 0; br < 2; ++br) {
    // layer 1: h = lrelu(x @ w1 + b1)
    v8f acc1[4] = {};
    wave_gemm_g<Dc>((gcp)(x + (size_t)row0 * Dc), Dc, (gcp)W1[br], acc1);
    bias_act(acc1, (gcp)B1[br], true);
    __syncthreads();
    store_tile_l(wt, LDT, acc1);
    __syncthreads();
    // layer 2: e = h @ w2 + b2
    v8f acc2[4] = {};
    wave_gemm_l<Hc>((lcp)wt, LDT, (gcp)W2[br], acc2);
    bias_act(acc2, (gcp)B2[br], false);
    store_tile_g((gp)(EO[br] + (size_t)row0 * Hc), Hc, acc2);
    __syncthreads();
    store_tile_l(wt, LDT, acc2);
    __syncthreads();
    // score projection: u = self_e @ w1_s ; v = nb_e @ w1_n + comb_b1
    v8f acc3[4] = {};
    wave_gemm_l<Hc>((lcp)wt, LDT, (gcp)CW[br], acc3);
    if (br == 1) bias_act(acc3, (gcp)comb_b1, false);
    store_tile_g((gp)(UV[br] + (size_t)row0 * Hc), Hc, acc3);
    __syncthreads();
  }
}

// -------- Kernel B: scores + mask + softmax + WMMA aggregation (fused) -------
__global__ void __launch_bounds__(256)
gat_attn_kernel(const int* __restrict__ edges,
                const float* __restrict__ comb_w2, const float* __restrict__ comb_b2,
                const float* __restrict__ self_e, const float* __restrict__ nb_e,
                const float* __restrict__ u, const float* __restrict__ v,
                float* __restrict__ out) {
  __shared__ __align__(16) float su[16 * Hc];   // u tile for 16 i-rows
  __shared__ __align__(16) float sc[16 * Nc];   // scores -> weights
  __shared__ __align__(16) float w2s[Hc];
  __shared__ int hasnb[16];

  const int tid = threadIdx.x;
  const int b   = blockIdx.x >> 5;              // 32 i-tiles per batch
  const int i0  = (blockIdx.x & 31) * 16;

  // stage 1: async global->LDS staging of u tile (4KB) and comb_w2 (256B)
  {
    const float* usrc = u + ((size_t)(b * Nc + i0)) * Hc;
    const unsigned voff = (unsigned)tid * 16u;
    const unsigned ldsa = (unsigned)(size_t)(lp)su + voff;
    asm volatile("global_load_async_to_lds_b128 %0, %1, %2 offset:0"
                 :: "v"(ldsa), "v"(voff), "s"(usrc) : "memory");
    if (tid < 16) {
      const unsigned ldsw = (unsigned)(size_t)(lp)w2s + voff;
      asm volatile("global_load_async_to_lds_b128 %0, %1, %2 offset:0"
                   :: "v"(ldsw), "v"(voff), "s"(comb_w2) : "memory");
    }
  }
  // warm caches for the stage-4 WMMA B-operand (nb_e slice, L2-resident)
  const float* nbb_pf = nb_e + (size_t)b * Nc * Hc;
  for (int off = tid * 32; off < Nc * Hc; off += 256 * 32)
    __builtin_prefetch(nbb_pf + off, 0, 3);

  asm volatile("s_wait_asynccnt 0x0" ::: "memory");
  __syncthreads();

  const float bias2 = *(gcp)comb_b2;

  // stage 2: scores[it][j] = sum_k lrelu(u[i,k] + v[j,k]) * w2[k] + b2, masked
  for (int j = tid; j < Nc; j += 256) {
    gcp vrow = (gcp)(v + ((size_t)(b * Nc + j)) * Hc);
    float acc[16];
#pragma unroll
    for (int it = 0; it < 16; ++it) acc[it] = bias2;
    for (int k = 0; k < Hc; ++k) {
      const float vv = vrow[k];
      const float wk = w2s[k];
#pragma unroll
      for (int it = 0; it < 16; ++it)
        acc[it] += lrelu(su[it * Hc + k] + vv) * wk;
    }
    gip erow = (gip)(edges + (size_t)b * Nc * Nc + (size_t)j * Nc + i0);
#pragma unroll
    for (int it = 0; it < 16; ++it) {
      const int irow = i0 + it;
      const bool m = (erow[it] != 0) && (j != irow);   // edges[b,j,i] && i!=j
      sc[it * Nc + j] = m ? acc[it] : kNeg;
    }
  }
  __syncthreads();

  // stage 3: per-row softmax (masked entries underflow exp -> exactly 0)
  const int wave = tid >> 5;
  const int lane = tid & 31;
  for (int it = wave * 2; it < wave * 2 + 2; ++it) {
    float mx = kNeg;
    for (int j = lane; j < Nc; j += 32) mx = fmaxf(mx, sc[it * Nc + j]);
#pragma unroll
    for (int o = 16; o > 0; o >>= 1) mx = fmaxf(mx, __shfl_xor(mx, o, 32));
    float sum = 0.0f;
    for (int j = lane; j < Nc; j += 32) {
      float e = __expf(sc[it * Nc + j] - mx);
      sc[it * Nc + j] = e;
      sum += e;
    }
#pragma unroll
    for (int o = 16; o > 0; o >>= 1) sum += __shfl_xor(sum, o, 32);
    const float inv = 1.0f / sum;
    for (int j = lane; j < Nc; j += 32) sc[it * Nc + j] *= inv;
    if (lane == 0) hasnb[it] = (mx > -1.0e30f) ? 1 : 0;
  }
  __syncthreads();

  // stage 4: agg(16x64) = weights(16x512) @ nb_e(512x64) via WMMA; + self_e
  if (wave < 4) {
    const int nt   = wave;
    const int l15  = lane & 15;
    const int koff = (lane >> 4) * 2;
    const int n    = nt * 16 + l15;
    gcp nbb = (gcp)(nb_e + (size_t)b * Nc * Hc);
    lcp scp = (lcp)sc;
    v8f acc = {};
#pragma unroll 4
    for (int k0 = 0; k0 < Nc; k0 += 4) {
      const int ka = k0 + koff;
      v2f a = *(lv2)&scp[l15 * Nc + ka];          // ds_load_b64
      v2f bf;
      bf.x = nbb[ka * Hc + n];
      bf.y = nbb[(ka + 1) * Hc + n];
      acc = __builtin_amdgcn_wmma_f32_16x16x4_f32(false, a, false, bf,
                                                  (short)0, acc, false, false);
    }
    const int mbase = (lane >> 4) * 8;
    gcp sep = (gcp)self_e;
    gp  outg = (gp)out;
#pragma unroll
    for (int r = 0; r < 8; ++r) {
      const int m = mbase + r;
      const size_t row = (size_t)(b * Nc + i0 + m);
      const float se = sep[row * Hc + n];
      outg[row * Hc + n] = hasnb[m] ? (acc[r] + se) : 0.0f;
    }
  }
}

extern "C" void kernel_launch(void* const* d_in, const int* in_sizes, int n_in,
                              void* d_out, int out_size, void* d_ws, size_t ws_size,
                              hipStream_t stream) {
  const float* nodes   = (const float*)d_in[0];
  const int*   edges   = (const int*)d_in[1];
  const float* self_w1 = (const float*)d_in[2];
  const float* self_b1 = (const float*)d_in[3];
  const float* self_w2 = (const float*)d_in[4];
  const float* self_b2 = (const float*)d_in[5];
  const float* nb_w1   = (const float*)d_in[6];
  const float* nb_b1   = (const float*)d_in[7];
  const float* nb_w2   = (const float*)d_in[8];
  const float* nb_b2   = (const float*)d_in[9];
  const float* comb_w1 = (const float*)d_in[10];
  const float* comb_b1 = (const float*)d_in[11];
  const float* comb_w2 = (const float*)d_in[12];
  const float* comb_b2 = (const float*)d_in[13];

  float* ws = (float*)d_ws;
  const size_t stride = (size_t)Bc * Nc * Hc;   // 262144 floats
  float* self_e = ws;
  float* nb_e   = ws + stride;
  float* u      = ws + 2 * stride;
  float* v      = ws + 3 * stride;

  gat_mlp_kernel<<<dim3((Bc * Nc / 16) / 8), dim3(256), 0, stream>>>(
      nodes, self_w1, self_b1, self_w2, self_b2,
      nb_w1, nb_b1, nb_w2, nb_b2, comb_w1, comb_b1,
      self_e, nb_e, u, v);

  gat_attn_kernel<<<dim3(Bc * (Nc / 16)), dim3(256), 0, stream>>>(
      edges, comb_w2, comb_b2, self_e, nb_e, u, v, (float*)d_out);
}